// AdditiveAttention_19911468385063
// MI455X (gfx1250) — compile-verified
//
#include <hip/hip_runtime.h>
#include <hip/hip_bf16.h>

typedef float v2f __attribute__((ext_vector_type(2)));
typedef float v8f __attribute__((ext_vector_type(8)));

// Problem dimensions (compile-time: fold all strides into immediate offsets).
constexpr int B_  = 16;
constexpr int NQ  = 128;
constexpr int NK  = 512;
constexpr int D_  = 256;   // DQ == DK
constexpr int H_  = 128;
constexpr int DV  = 256;

// ---------------------------------------------------------------------------
// fast tanh pair: CDNA5 V_TANH_F32 transcendental.
// TRANS-op hazard rule (ISA 7.4): 1 independent op or V_NOP after a trans op
// before its output is consumed. Two independent tanhs cover each other for
// the first slot; trailing v_nop covers the second.
// ---------------------------------------------------------------------------
__device__ __forceinline__ void tanh2(float x0, float x1, float& y0, float& y1) {
#if __has_builtin(__builtin_amdgcn_tanhf)
    y0 = __builtin_amdgcn_tanhf(x0);
    y1 = __builtin_amdgcn_tanhf(x1);
#else
    asm volatile("v_tanh_f32 %0, %2\n\t"
                 "v_tanh_f32 %1, %3\n\t"
                 "v_nop"
                 : "=&v"(y0), "=&v"(y1)
                 : "v"(x0), "v"(x1));
#endif
}

// ---------------------------------------------------------------------------
// Projection: out = X (MxD) * W^T, W is (H x D) row-major. One wave per 16x16
// output tile; v_wmma_f32_16x16x4_f32 with TWO interleaved accumulator chains
// (even/odd K-blocks) to break the WMMA RAW chain on the matrix pipe.
// A frag (16x4 f32): lane<16 -> row=lane, {K0,K1}; lane>=16 -> row, {K2,K3}
// B frag (4x16 f32): col N = lane&15; vgpr0/1 hold K = 2*half + {0,1}
// TRANSPOSE_OUT=true writes out[h*M + m] (KpT[b][h][k] layout for keys).
// ---------------------------------------------------------------------------
template <int M, bool TRANSPOSE_OUT>
__global__ __launch_bounds__(32)
void proj_wmma_kernel(const float* __restrict__ X, const float* __restrict__ W,
                      float* __restrict__ out) {
    const int b    = blockIdx.y;
    const int tm   = blockIdx.x / (H_ / 16);
    const int th   = blockIdx.x % (H_ / 16);
    const int lane = threadIdx.x;
    const int half = lane >> 4;   // 0 or 1
    const int l16  = lane & 15;
    const int m0   = tm << 4, h0 = th << 4;

    const float* xrow = X + b * (M * D_) + (m0 + l16) * D_ + 2 * half;
    const float* wrow = W + (h0 + l16) * D_ + 2 * half;

    v8f c0 = {}, c1 = {};
#pragma unroll 4
    for (int d = 0; d < D_; d += 8) {
        v2f a0, b0, a1, b1;
        a0.x = xrow[d + 0]; a0.y = xrow[d + 1];
        b0.x = wrow[d + 0]; b0.y = wrow[d + 1];
        a1.x = xrow[d + 4]; a1.y = xrow[d + 5];
        b1.x = wrow[d + 4]; b1.y = wrow[d + 5];
        c0 = __builtin_amdgcn_wmma_f32_16x16x4_f32(false, a0, false, b0,
                                                   (short)0, c0, false, false);
        c1 = __builtin_amdgcn_wmma_f32_16x16x4_f32(false, a1, false, b1,
                                                   (short)0, c1, false, false);
    }
    v8f c = c0 + c1;

    float* outb = out + b * (M * H_);
#pragma unroll
    for (int r = 0; r < 8; ++r) {
        const int row = m0 + r + 8 * half;   // C/D layout: vgpr r -> M=r / M=r+8
        const int col = h0 + l16;
        if (TRANSPOSE_OUT)
            outb[col * M + row] = c[r];
        else
            outb[row * H_ + col] = c[r];
    }
}

// ---------------------------------------------------------------------------
// scores[b,q,k] = sum_h wv[h] * tanh(Qp[b,q,h] + KpT[b,h,k])
// One thread per k-column, QPB query rows amortized per KpT load.
// Qp/wv accesses are wave-uniform -> SGPR loads; KpT loads are coalesced with
// compile-time strides (immediate offsets inside load clauses).
// ---------------------------------------------------------------------------
#define QPB 8
__global__ __launch_bounds__(256)
void score_kernel(const float* __restrict__ Qp, const float* __restrict__ KpT,
                  const float* __restrict__ wv, float* __restrict__ scores) {
    const int b  = blockIdx.z;
    const int q0 = blockIdx.y * QPB;
    const int k  = blockIdx.x * 256 + threadIdx.x;

    const float* kcol = KpT + b * (H_ * NK) + k;
    const float* qrow = Qp + (b * NQ + q0) * H_;

    float acc[QPB];
#pragma unroll
    for (int j = 0; j < QPB; ++j) acc[j] = 0.f;

#pragma unroll 2
    for (int h = 0; h < H_; ++h) {
        const float kv = kcol[h * NK];   // coalesced vector load, imm offset
        const float w  = wv[h];          // uniform -> SGPR
        float x[QPB], t[QPB];
#pragma unroll
        for (int j = 0; j < QPB; ++j) x[j] = qrow[j * H_ + h] + kv;
#pragma unroll
        for (int j = 0; j < QPB; j += 2) tanh2(x[j], x[j + 1], t[j], t[j + 1]);
#pragma unroll
        for (int j = 0; j < QPB; ++j) acc[j] = fmaf(w, t[j], acc[j]);
    }

#pragma unroll
    for (int j = 0; j < QPB; ++j)
        scores[(b * NQ + q0 + j) * NK + k] = acc[j];
}

// ---------------------------------------------------------------------------
// Softmax over the QUERY axis (reference: softmax(scores, axis=1)); one thread
// per (b,k) column, in-place. Consecutive threads -> consecutive addresses.
// ---------------------------------------------------------------------------
__global__ __launch_bounds__(256)
void softmax_q_kernel(float* __restrict__ s) {
    const int b = blockIdx.y;
    const int k = blockIdx.x * 256 + threadIdx.x;
    float* col = s + b * (NQ * NK) + k;

    float m = -3.0e38f;
#pragma unroll 4
    for (int q = 0; q < NQ; ++q) m = fmaxf(m, col[q * NK]);
    float sum = 0.f;
#pragma unroll 4
    for (int q = 0; q < NQ; ++q) sum += __expf(col[q * NK] - m);
    const float inv = 1.0f / sum;
#pragma unroll 4
    for (int q = 0; q < NQ; ++q)
        col[q * NK] = __expf(col[q * NK] - m) * inv;
}

// ---------------------------------------------------------------------------
// Batched GEMM: out[b] (NQ x DV) = attn[b] (NQ x NK) * values[b] (NK x DV),
// all row-major. One wave per 16x16 tile; two interleaved WMMA chains.
// ---------------------------------------------------------------------------
__global__ __launch_bounds__(32)
void bmm_wmma_kernel(const float* __restrict__ A, const float* __restrict__ Bm,
                     float* __restrict__ C) {
    const int b    = blockIdx.y;
    const int tm   = blockIdx.x / (DV / 16);
    const int tn   = blockIdx.x % (DV / 16);
    const int lane = threadIdx.x;
    const int half = lane >> 4;
    const int l16  = lane & 15;
    const int m0   = tm << 4, n0 = tn << 4;

    const float* Ab = A + b * (NQ * NK) + (m0 + l16) * NK + 2 * half;
    const float* Bb = Bm + b * (NK * DV) + (2 * half) * DV + n0 + l16;

    v8f c0 = {}, c1 = {};
#pragma unroll 4
    for (int k = 0; k < NK; k += 8) {
        v2f a0, b0, a1, b1;
        a0.x = Ab[k + 0];
        a0.y = Ab[k + 1];
        b0.x = Bb[(k + 0) * DV];          // row k + 2*half
        b0.y = Bb[(k + 1) * DV];          // row k + 2*half + 1
        a1.x = Ab[k + 4];
        a1.y = Ab[k + 5];
        b1.x = Bb[(k + 4) * DV];
        b1.y = Bb[(k + 5) * DV];
        c0 = __builtin_amdgcn_wmma_f32_16x16x4_f32(false, a0, false, b0,
                                                   (short)0, c0, false, false);
        c1 = __builtin_amdgcn_wmma_f32_16x16x4_f32(false, a1, false, b1,
                                                   (short)0, c1, false, false);
    }
    v8f c = c0 + c1;

    float* Cb = C + b * (NQ * DV);
#pragma unroll
    for (int r = 0; r < 8; ++r)
        Cb[(m0 + r + 8 * half) * DV + n0 + l16] = c[r];
}

// ---------------------------------------------------------------------------
extern "C" void kernel_launch(void* const* d_in, const int* in_sizes, int n_in,
                              void* d_out, int out_size, void* d_ws, size_t ws_size,
                              hipStream_t stream) {
    (void)in_sizes; (void)n_in; (void)out_size; (void)ws_size;

    const float* queries = (const float*)d_in[0];
    const float* keys    = (const float*)d_in[1];
    const float* values  = (const float*)d_in[2];
    const float* Wq      = (const float*)d_in[3];
    const float* Wk      = (const float*)d_in[4];
    const float* wv      = (const float*)d_in[5];
    float*       out     = (float*)d_out;

    float* Qp     = (float*)d_ws;                     // B*NQ*H   (1 MB)
    float* KpT    = Qp + (size_t)B_ * NQ * H_;        // B*H*NK   (4 MB), transposed
    float* scores = KpT + (size_t)B_ * H_ * NK;       // B*NQ*NK  (4 MB), becomes attn in-place

    // 1) q projection: Qp[b,q,h]
    proj_wmma_kernel<NQ, false><<<dim3((NQ / 16) * (H_ / 16), B_), 32, 0, stream>>>(
        queries, Wq, Qp);
    // 2) k projection, stored transposed: KpT[b,h,k]
    proj_wmma_kernel<NK, true><<<dim3((NK / 16) * (H_ / 16), B_), 32, 0, stream>>>(
        keys, Wk, KpT);
    // 3) tanh feature + wv reduction (dominant stage, VALU/transcendental)
    score_kernel<<<dim3(NK / 256, NQ / QPB, B_), 256, 0, stream>>>(
        Qp, KpT, wv, scores);
    // 4) softmax over query axis (in-place -> attn)
    softmax_q_kernel<<<dim3(NK / 256, B_), 256, 0, stream>>>(scores);
    // 5) out[b] = attn[b] (NQ x NK) * values[b] (NK x DV)
    bmm_wmma_kernel<<<dim3((NQ / 16) * (DV / 16), B_), 32, 0, stream>>>(
        scores, values, out);
}